// BLAMem_24704651886867
// MI455X (gfx1250) — compile-verified
//
#include <hip/hip_runtime.h>

// ---------------------------------------------------------------------------
// BLAMem depth-3 log-signature pipeline for MI455X (gfx1250, wave32).
//   K1: per-chunk log-signature (8192 waves), level-3 via V_WMMA_F32_16X16X4_F32
//   K2: BCH tree-fold, 9 rounds; operands staged with GLOBAL_LOAD_ASYNC_TO_LDS,
//       level-3 terms batched into two small f32 WMMA GEMMs
//   K3: MLP head (16x819 @ 819x256) via f32 WMMA, then 256->1 reduction
// ---------------------------------------------------------------------------

typedef float v2f __attribute__((ext_vector_type(2)));
typedef float v8f __attribute__((ext_vector_type(8)));

static constexpr int SLOT   = 832;    // 819 payload, padded (16B-aligned slots)
static constexpr int NCHUNK = 512;    // chunks per batch (16384/32)
static constexpr int NB     = 16;
static constexpr int CIN    = 8;

__device__ __forceinline__ v8f wmma4(v2f a, v2f b, v8f c) {
  // D = A(16x4 f32) * B(4x16 f32) + C(16x16 f32)
  return __builtin_amdgcn_wmma_f32_16x16x4_f32(false, a, false, b, (short)0, c,
                                               false, false);
}

// Async global->LDS copy (ASYNCcnt path). lds_off is the wave-relative LDS
// byte address == low 32 bits of the flat pointer to __shared__ (ISA 10.2).
__device__ __forceinline__ void async_g2l_b128(unsigned lds_off, const float* g) {
  asm volatile("global_load_async_to_lds_b128 %0, %1, off"
               :: "v"(lds_off), "v"(g) : "memory");
}
__device__ __forceinline__ void wait_async0() {
  asm volatile("s_wait_asynccnt 0" ::: "memory");
}

// ---------------------------------------------------------------------------
// Kernel 1: chunk log-signature. One wave per (batch, chunk). lane == step s.
// ---------------------------------------------------------------------------
__launch_bounds__(32)
__global__ void k_chunk_logsig(const float* __restrict__ x,
                               float* __restrict__ out) {
  __shared__ float Mlds[96 * 33];   // A-matrix [ij][s], padded rows/cols
  __shared__ float Blds[32 * 16];   // B-matrix [s][k], cols 9..15 zero
  __shared__ float dxS[32 * 9];
  __shared__ float PS[32 * 9];
  __shared__ float S1lds[9];
  __shared__ float S2lds[81];
  __shared__ float S3lds[96 * 16];

  const int lane = threadIdx.x;
  const int bc   = blockIdx.x;                   // b*512 + chunk

  for (int t = lane; t < 96 * 33; t += 32) Mlds[t] = 0.0f;
  for (int t = lane; t < 32 * 16; t += 32) Blds[t] = 0.0f;

  // increments (per-chunk basepoint: zeros with t=0)
  const float* rowp = x + ((size_t)bc * 32 + lane) * CIN;
  float dx[9];
  {
    float cur[CIN], prv[CIN];
#pragma unroll
    for (int c = 0; c < CIN; ++c) cur[c] = rowp[c];
    if (lane == 0) {
#pragma unroll
      for (int c = 0; c < CIN; ++c) prv[c] = 0.0f;
    } else {
      const float* rp = rowp - CIN;
#pragma unroll
      for (int c = 0; c < CIN; ++c) prv[c] = rp[c];
    }
#pragma unroll
    for (int c = 0; c < CIN; ++c) dx[c] = cur[c] - prv[c];
    dx[8] = 1.0f / 32.0f;                        // time channel increment
  }
  __builtin_prefetch(rowp + 32 * CIN, 0, 1);     // gfx1250 global_prefetch_b8

  // level-1 inclusive scan across lanes (steps); P = exclusive prefix
#pragma unroll
  for (int c = 0; c < 9; ++c) {
    float incl = dx[c];
#pragma unroll
    for (int off = 1; off < 32; off <<= 1) {
      float t = __shfl_up(incl, off, 32);
      if (lane >= off) incl += t;
    }
    dxS[lane * 9 + c] = dx[c];
    PS[lane * 9 + c]  = incl - dx[c];
    Blds[lane * 16 + c] = dx[c];
    if (lane == 31) S1lds[c] = incl;
  }

  // level-2 scan per (i,j); build A-matrix M[s,ij] for the level-3 GEMM:
  // M = S2e + 0.5*P_i*dx_j + dx_i*dx_j/6  (shared trailing (x) dx_k)
  for (int ij = 0; ij < 81; ++ij) {
    const int i = ij / 9, j = ij % 9;
    const float di = dxS[lane * 9 + i];
    const float dj = dxS[lane * 9 + j];
    const float pi = PS[lane * 9 + i];
    const float st2 = pi * dj + 0.5f * di * dj;
    float incl = st2;
#pragma unroll
    for (int off = 1; off < 32; off <<= 1) {
      float t = __shfl_up(incl, off, 32);
      if (lane >= off) incl += t;
    }
    Mlds[ij * 33 + lane] = (incl - st2) + 0.5f * pi * dj + di * dj * (1.0f / 6.0f);
    if (lane == 31) S2lds[ij] = incl;
  }
  __syncthreads();

  // S3[81x9] = M[81x32] * dx[32x9] : 6 M-tiles x 8 K-steps of f32 WMMA
  const int half = lane >> 4;
  const int l16  = lane & 15;
  v8f acc[6];
  const v8f vz = {};
#pragma unroll
  for (int m = 0; m < 6; ++m) acc[m] = vz;
#pragma unroll
  for (int kk = 0; kk < 8; ++kk) {
    const int kb = kk * 4 + half * 2;
    v2f bf;
    bf.x = Blds[kb * 16 + l16];
    bf.y = Blds[(kb + 1) * 16 + l16];
#pragma unroll
    for (int m = 0; m < 6; ++m) {
      const int r = m * 16 + l16;
      v2f af;
      af.x = Mlds[r * 33 + kb];
      af.y = Mlds[r * 33 + kb + 1];
      acc[m] = wmma4(af, bf, acc[m]);
    }
  }
#pragma unroll
  for (int m = 0; m < 6; ++m)
#pragma unroll
    for (int v = 0; v < 8; ++v)
      S3lds[(m * 16 + v + half * 8) * 16 + l16] = acc[m][v];
  __syncthreads();

  // log3 and emit the 819-float slot
  float* slot = out + (size_t)bc * SLOT;
  if (lane < 9) slot[lane] = S1lds[lane];
  for (int idx = lane; idx < 81; idx += 32) {
    const int i = idx / 9, j = idx % 9;
    slot[9 + idx] = S2lds[idx] - 0.5f * S1lds[i] * S1lds[j];
  }
  for (int idx = lane; idx < 729; idx += 32) {
    const int i = idx / 81, jk = idx % 81;
    const int ij = idx / 9, k = idx % 9;
    const int j = ij % 9;
    slot[90 + idx] = S3lds[ij * 16 + k]
                   - 0.5f * (S1lds[i] * S2lds[jk] + S2lds[ij] * S1lds[k])
                   + S1lds[i] * S1lds[j] * S1lds[k] * (1.0f / 3.0f);
  }
}

// ---------------------------------------------------------------------------
// Kernel 2: one wave per BCH merge c = log(exp(a)*exp(b)) truncated at 3.
// Level-3: c3 = a3+b3 + [7 u(x)V terms as K=8 GEMM] + [4 V(x)u terms as K=4].
// ---------------------------------------------------------------------------
__launch_bounds__(32)
__global__ void k_bch(const float* __restrict__ in, float* __restrict__ out,
                      int m_out, int in_stride, int out_stride) {
  __shared__ __align__(16) float aL[SLOT];
  __shared__ __align__(16) float bL[SLOT];
  __shared__ float A11[81], B11[81], S11[81], E2[81], F2[81], SM2[81];
  __shared__ float U[16 * 8];       // thin operand for the u(x)V GEMM
  __shared__ float M1[16 * 96];     // u(x)V result  [i][jk]
  __shared__ float M2[96 * 16];     // V(x)u result  [ij][k]

  const int lane = threadIdx.x;
  const int gid  = blockIdx.x;
  const int b    = gid / m_out;
  const int i    = gid - b * m_out;
  const float* ap = in + ((size_t)b * in_stride + 2 * i) * SLOT;
  const float* bp = ap + SLOT;
  float* op = out + ((size_t)b * out_stride + i) * SLOT;

  // Stage both operands with the CDNA5 async global->LDS path (ASYNCcnt).
  // 205 x B128 covers floats [0,820) of each 832-float slot; both sides 16B
  // aligned. LDS byte address = low 32 bits of the flat __shared__ pointer.
  {
    const unsigned aoff = (unsigned)(unsigned long long)(uintptr_t)&aL[0];
    const unsigned boff = (unsigned)(unsigned long long)(uintptr_t)&bL[0];
    for (int q = lane; q < 205; q += 32) {
      async_g2l_b128(aoff + (unsigned)q * 16u, ap + q * 4);
      async_g2l_b128(boff + (unsigned)q * 16u, bp + q * 4);
    }
    wait_async0();
  }
  __syncthreads();

  // 81-sized intermediates + c1/c2 outputs
  for (int idx = lane; idx < 81; idx += 32) {
    const int ii = idx / 9, jj = idx % 9;
    const float a1i = aL[ii], a1j = aL[jj];
    const float b1i = bL[ii], b1j = bL[jj];
    const float s1i = a1i + b1i, s1j = a1j + b1j;
    const float a11 = a1i * a1j, b11 = b1i * b1j;
    const float e2 = aL[9 + idx] + 0.5f * a11;
    const float f2 = bL[9 + idx] + 0.5f * b11;
    A11[idx] = a11; B11[idx] = b11; S11[idx] = s1i * s1j;
    E2[idx] = e2;   F2[idx] = f2;
    SM2[idx] = e2 + f2 + a1i * b1j;
    op[9 + idx] = aL[9 + idx] + bL[9 + idx] + 0.5f * (a1i * b1j - b1i * a1j);
  }
  if (lane < 9) op[lane] = aL[lane] + bL[lane];
  if (lane < 16) {
    float u[8] = {0, 0, 0, 0, 0, 0, 0, 0};
    if (lane < 9) {
      const float a1 = aL[lane], b1 = bL[lane], s1 = a1 + b1;
      u[0] = 0.5f * a1;  u[1] = a1 * (1.0f / 6.0f); u[2] = a1;
      u[3] = 0.5f * b1;  u[4] = b1 * (1.0f / 6.0f);
      u[5] = -0.5f * s1; u[6] = s1 * (1.0f / 3.0f);
    }
#pragma unroll
    for (int t = 0; t < 8; ++t) U[lane * 8 + t] = u[t];
  }
  __syncthreads();

  const int half = lane >> 4;
  const int l16  = lane & 15;
  const v8f vz = {};

  // GEMM 1 (u(x)V): M1[9x81] = U[9x8] * V[8x81]
  auto vrow1 = [&](int t, int col) -> float {
    if (col >= 81) return 0.0f;
    switch (t) {
      case 0: return aL[9 + col];   // a2        (0.5 folded in U)
      case 1: return A11[col];      // a1(x)a1   (1/6)
      case 2: return F2[col];       // b2+0.5b1b1 (1.0)
      case 3: return bL[9 + col];   // b2        (0.5)
      case 4: return B11[col];      // b1(x)b1   (1/6)
      case 5: return SM2[col];      // mul3 s2   (-0.5)
      case 6: return S11[col];      // s1(x)s1   (1/3)
      default: return 0.0f;
    }
  };
  v8f acc1[6];
#pragma unroll
  for (int nn = 0; nn < 6; ++nn) acc1[nn] = vz;
#pragma unroll
  for (int kk = 0; kk < 2; ++kk) {
    const int kb = kk * 4 + half * 2;
    v2f af;
    af.x = U[l16 * 8 + kb];
    af.y = U[l16 * 8 + kb + 1];
#pragma unroll
    for (int nn = 0; nn < 6; ++nn) {
      const int col = nn * 16 + l16;
      v2f bf;
      bf.x = vrow1(kb, col);
      bf.y = vrow1(kb + 1, col);
      acc1[nn] = wmma4(af, bf, acc1[nn]);
    }
  }
#pragma unroll
  for (int nn = 0; nn < 6; ++nn)
#pragma unroll
    for (int v = 0; v < 8; ++v)
      M1[(v + half * 8) * 96 + nn * 16 + l16] = acc1[nn][v];

  // GEMM 2 (V(x)u): M2[81x9] = V[81x4] * Ub[4x9]
  auto varow2 = [&](int t, int r) -> float {
    if (r >= 81) return 0.0f;
    switch (t) {
      case 0: return aL[9 + r];     // a2   (0.5 folded into Ub)
      case 1: return bL[9 + r];     // b2   (0.5)
      case 2: return E2[r];         // a2+0.5a1a1 (1.0, (x) b1)
      case 3: return SM2[r];        // mul3 s2    (-0.5, (x) s1)
      default: return 0.0f;
    }
  };
  auto ubrow2 = [&](int t, int k) -> float {
    if (k >= 9) return 0.0f;
    switch (t) {
      case 0: return 0.5f * aL[k];
      case 1: return 0.5f * bL[k];
      case 2: return bL[k];
      case 3: return -0.5f * (aL[k] + bL[k]);
      default: return 0.0f;
    }
  };
  const int kb2 = half * 2;
  v2f bf2;
  bf2.x = ubrow2(kb2, l16);
  bf2.y = ubrow2(kb2 + 1, l16);
  v8f acc2[6];
#pragma unroll
  for (int m = 0; m < 6; ++m) acc2[m] = vz;
#pragma unroll
  for (int m = 0; m < 6; ++m) {
    const int r = m * 16 + l16;
    v2f af2;
    af2.x = varow2(kb2, r);
    af2.y = varow2(kb2 + 1, r);
    acc2[m] = wmma4(af2, bf2, acc2[m]);
  }
#pragma unroll
  for (int m = 0; m < 6; ++m)
#pragma unroll
    for (int v = 0; v < 8; ++v)
      M2[(m * 16 + v + half * 8) * 16 + l16] = acc2[m][v];
  __syncthreads();

  for (int idx = lane; idx < 729; idx += 32) {
    const int ii = idx / 81, jk = idx % 81;
    const int ij = idx / 9,  k  = idx % 9;
    op[90 + idx] = aL[90 + idx] + bL[90 + idx] + M1[ii * 96 + jk] + M2[ij * 16 + k];
  }
}

// ---------------------------------------------------------------------------
// Kernel 3: MLP head. h = relu(flat @ W1 + b1); out = h @ W2 + b2.
// One 256-thread block (8 waves); each wave owns 2 N-tiles of 16 columns.
// ---------------------------------------------------------------------------
__launch_bounds__(256)
__global__ void k_mlp(const float* __restrict__ fin, long fstride,
                      const float* __restrict__ W1, const float* __restrict__ b1v,
                      const float* __restrict__ W2, const float* __restrict__ b2v,
                      float* __restrict__ outv) {
  __shared__ float hw[16 * 256];
  const int tid  = threadIdx.x;
  const int lane = tid & 31;
  const int wave = tid >> 5;
  const int half = lane >> 4;
  const int l16  = lane & 15;
  const v8f vz = {};
  v8f acc[2] = {vz, vz};
  const float* arow = fin + (size_t)l16 * fstride;   // batch row l16
  for (int kk = 0; kk < 205; ++kk) {                 // K = 819 padded to 820
    const int kb = kk * 4 + half * 2;
    v2f af;
    af.x = (kb < 819) ? arow[kb] : 0.0f;
    af.y = (kb + 1 < 819) ? arow[kb + 1] : 0.0f;
#pragma unroll
    for (int nt = 0; nt < 2; ++nt) {
      const int col = (wave * 2 + nt) * 16 + l16;
      v2f bf;
      bf.x = (kb < 819) ? W1[(size_t)kb * 256 + col] : 0.0f;
      bf.y = (kb + 1 < 819) ? W1[(size_t)(kb + 1) * 256 + col] : 0.0f;
      acc[nt] = wmma4(af, bf, acc[nt]);
    }
  }
#pragma unroll
  for (int nt = 0; nt < 2; ++nt)
#pragma unroll
    for (int v = 0; v < 8; ++v) {
      const int rowb = v + half * 8;
      const int col  = (wave * 2 + nt) * 16 + l16;
      float h = acc[nt][v] + b1v[col];
      h = (h > 0.0f) ? h : 0.0f;
      hw[rowb * 256 + col] = h * W2[col];
    }
  __syncthreads();
  if (tid < 16) {
    float s = b2v[0];
    for (int c = 0; c < 256; ++c) s += hw[tid * 256 + c];
    outv[tid] = s;
  }
}

// ---------------------------------------------------------------------------
extern "C" void kernel_launch(void* const* d_in, const int* in_sizes, int n_in,
                              void* d_out, int out_size, void* d_ws, size_t ws_size,
                              hipStream_t stream) {
  (void)in_sizes; (void)n_in; (void)out_size; (void)ws_size;
  const float* x  = (const float*)d_in[0];
  const float* W1 = (const float*)d_in[1];
  const float* b1 = (const float*)d_in[2];
  const float* W2 = (const float*)d_in[3];
  const float* b2 = (const float*)d_in[4];
  float* out = (float*)d_out;

  float* buf0 = (float*)d_ws;                              // 16*512 slots
  float* buf1 = buf0 + (size_t)NB * NCHUNK * SLOT;         // 16*256 slots

  // 1) per-chunk log-signatures
  k_chunk_logsig<<<dim3(NB * NCHUNK), dim3(32), 0, stream>>>(x, buf0);

  // 2) BCH tree-fold (associative): 9 ping-pong rounds, last result in buf1
  int m_in = NCHUNK;
  for (int r = 0; r < 9; ++r) {
    const int m_out = m_in >> 1;
    const float* inb;
    float* outb;
    int is, os;
    if ((r & 1) == 0) { inb = buf0; is = NCHUNK;     outb = buf1; os = NCHUNK / 2; }
    else              { inb = buf1; is = NCHUNK / 2; outb = buf0; os = NCHUNK; }
    k_bch<<<dim3(NB * m_out), dim3(32), 0, stream>>>(inb, outb, m_out, is, os);
    m_in = m_out;
  }

  // 3) MLP head on the 16 final 819-vectors
  k_mlp<<<dim3(1), dim3(256), 0, stream>>>(buf1, (long)(NCHUNK / 2) * SLOT,
                                           W1, b1, W2, b2, out);
}